// SRCCLoss_70798240907319
// MI455X (gfx1250) — compile-verified
//
#include <hip/hip_runtime.h>

// Problem geometry: pred/target (64, 8192) f32, viewed as (8192, 64).
#define M 64
#define ROWS 8192
#define NTOT (M * ROWS)          // 524288 per tensor
#define RED_BLOCKS 64
#define RED_THREADS 256
#define EPS 1e-8

typedef __attribute__((ext_vector_type(2))) float v2f;
typedef __attribute__((ext_vector_type(8))) float v8f;

// ---------------------------------------------------------------------------
// Kernel A: per-row soft_rank (L2 reg., strength 1) for one length-64 row.
// One 64-thread block per row. blocks [0,8192) -> pred, [8192,16384) -> target.
// ---------------------------------------------------------------------------
__global__ void softrank_kernel(const float* __restrict__ pred,
                                const float* __restrict__ target,
                                float* __restrict__ PR,
                                float* __restrict__ TR) {
    __shared__ float y[M];            // input row (theta)
    __shared__ float s[M];            // sorted descending
    __shared__ float zz[M];           // z = s - w
    __shared__ float S[M + 1];        // prefix sums of z
    __shared__ float Rm[M * (M + 1)]; // reverse-cummax matrix, padded stride 65
    __shared__ float pr[M];           // primal in sorted order

    const int c = threadIdx.x;
    const int blk = blockIdx.x;
    const float* xin;
    float* xout;
    int row;
    if (blk < ROWS) { xin = pred;   xout = PR; row = blk; }
    else            { xin = target; xout = TR; row = blk - ROWS; }

    // --- async global -> LDS row staging (gfx1250 async path) ---
    const float* src = xin + (size_t)row * M + c;
    unsigned ldsoff = (unsigned)(unsigned long long)(void*)&y[c];
    asm volatile("global_load_async_to_lds_b32 %0, %1, off"
                 :: "v"(ldsoff), "v"(src) : "memory");
    asm volatile("s_wait_asynccnt 0x0" ::: "memory");
    __syncthreads();

    // --- stable descending rank by counting (== argsort(-theta)) ---
    const float v = y[c];
    int rank = 0;
#pragma unroll 8
    for (int j = 0; j < M; ++j) {
        const float u = y[j];
        rank += ((u > v) || (u == v && j < c)) ? 1 : 0;
    }
    s[rank] = v;                      // rank is a permutation -> conflict-free
    __syncthreads();

    // --- z = s - w, w = [64..1] ; prefix sums ---
    const float sc = s[c];
    zz[c] = sc - (float)(M - c);
    __syncthreads();
    if (c == 0) {
        float acc = 0.0f;
        S[0] = 0.0f;
        for (int i = 0; i < M; ++i) { acc += zz[i]; S[i + 1] = acc; }
    }
    __syncthreads();

    // --- R[j][k] = max_{k'>=k} mean(z[j..k'])  (thread j = c) ---
    {
        const float Sj = S[c];
        float run = -3.4e38f;
        for (int k = M - 1; k >= c; --k) {
            const float mean = (S[k + 1] - Sj) / (float)(k - c + 1);
            run = fmaxf(run, mean);
            Rm[c * (M + 1) + k] = run;   // padded stride -> no bank conflicts
        }
    }
    __syncthreads();

    // --- dual_i = min_{j<=i} R[j][i]; primal = s - dual  (thread i = c) ---
    float d = 3.4e38f;
    for (int j = 0; j <= c; ++j) d = fminf(d, Rm[j * (M + 1) + c]);
    pr[c] = sc - d;
    __syncthreads();

    // --- unsort: out[c] = primal[inv[c]] = pr[rank_c] ---
    xout[(size_t)row * M + c] = pr[rank];
}

// ---------------------------------------------------------------------------
// Kernel B: 5 partial sums per block over PR/TR:
//   [sumP, sumT, sumP^2, sumT^2, sumP*T]   (fixed-order, deterministic)
// ---------------------------------------------------------------------------
__global__ void reduce_kernel(const float* __restrict__ PR,
                              const float* __restrict__ TR,
                              float* __restrict__ partials) {
    __shared__ float sm[RED_THREADS * 5];
    const int tid = blockIdx.x * blockDim.x + threadIdx.x;
    float s0 = 0.f, s1 = 0.f, s2 = 0.f, s3 = 0.f, s4 = 0.f;
    for (int i = tid; i < NTOT; i += RED_BLOCKS * RED_THREADS) {
        const float p = PR[i], t = TR[i];
        s0 += p; s1 += t; s2 += p * p; s3 += t * t; s4 += p * t;
    }
    sm[threadIdx.x * 5 + 0] = s0;
    sm[threadIdx.x * 5 + 1] = s1;
    sm[threadIdx.x * 5 + 2] = s2;
    sm[threadIdx.x * 5 + 3] = s3;
    sm[threadIdx.x * 5 + 4] = s4;
    __syncthreads();
    for (int off = RED_THREADS / 2; off > 0; off >>= 1) {
        if (threadIdx.x < off) {
#pragma unroll
            for (int k = 0; k < 5; ++k)
                sm[threadIdx.x * 5 + k] += sm[(threadIdx.x + off) * 5 + k];
        }
        __syncthreads();
    }
    if (threadIdx.x == 0) {
#pragma unroll
        for (int k = 0; k < 5; ++k)
            partials[blockIdx.x * 5 + k] = sm[k];
    }
}

// ---------------------------------------------------------------------------
// Kernel C: single wave. WMMA f32 16x16x4 column-sum of the 64x5 partials
// (A = ones(16x4), B = 4x16 chunks -> every C element = column sum), then the
// final scalar in f64. C layout: VGPR0, lanes 0..15 = C[0][N=lane].
// ---------------------------------------------------------------------------
__global__ void final_kernel(const float* __restrict__ partials,
                             float* __restrict__ out) {
    const int lane = threadIdx.x;         // 0..31, all active (EXEC all 1s)
    const int col = lane & 15;            // N index
    const int rowbase = (lane >> 4) << 1; // lanes 0-15 -> K{0,1}; 16-31 -> K{2,3}

    v2f a; a.x = 1.0f; a.y = 1.0f;        // A = ones(16x4)
    v8f c = {};
    const int cc = (col < 5) ? col : 0;
    const float msk = (col < 5) ? 1.0f : 0.0f;
#pragma unroll
    for (int t = 0; t < 16; ++t) {        // 16 chunks x K=4 = 64 partial rows
        const int r0 = t * 4 + rowbase;
        v2f b;
        b.x = partials[r0 * 5 + cc] * msk;
        b.y = partials[(r0 + 1) * 5 + cc] * msk;
        c = __builtin_amdgcn_wmma_f32_16x16x4_f32(
                /*neg_a=*/false, a, /*neg_b=*/false, b,
                /*c_mod=*/(short)0, c, /*reuse_a=*/false, /*reuse_b=*/false);
    }
    const float c0 = c[0];                // lane n (n<16): colsum[n]
    const float sp  = __shfl(c0, 0, 32);
    const float st  = __shfl(c0, 1, 32);
    const float spp = __shfl(c0, 2, 32);
    const float stt = __shfl(c0, 3, 32);
    const float spt = __shfl(c0, 4, 32);

    if (lane == 0) {
        const double Nd = (double)NTOT;
        const double mp = (double)sp / Nd;
        const double mt = (double)st / Nd;
        double varp = (double)spp - Nd * mp * mp;   // = sum (p - mp)^2
        double vart = (double)stt - Nd * mt * mt;
        const double cov = (double)spt - Nd * mp * mt;
        if (varp < 0.0) varp = 0.0;
        if (vart < 0.0) vart = 0.0;
        const double denom = (sqrt(varp) + EPS) * (sqrt(vart) + EPS);
        out[0] = (float)(1.0 - cov / denom);
    }
}

// ---------------------------------------------------------------------------
extern "C" void kernel_launch(void* const* d_in, const int* in_sizes, int n_in,
                              void* d_out, int out_size, void* d_ws, size_t ws_size,
                              hipStream_t stream) {
    const float* pred   = (const float*)d_in[0];
    const float* target = (const float*)d_in[1];

    float* PR       = (float*)d_ws;            // 524288 f32
    float* TR       = PR + NTOT;               // 524288 f32
    float* partials = TR + NTOT;               // 64*5 f32

    softrank_kernel<<<2 * ROWS, M, 0, stream>>>(pred, target, PR, TR);
    reduce_kernel<<<RED_BLOCKS, RED_THREADS, 0, stream>>>(PR, TR, partials);
    final_kernel<<<1, 32, 0, stream>>>(partials, (float*)d_out);
}